// SqueezeExcitation_68487548502226
// MI455X (gfx1250) — compile-verified
//
#include <hip/hip_runtime.h>
#include <hip/hip_bf16.h>

typedef __attribute__((ext_vector_type(2))) float v2f;
typedef __attribute__((ext_vector_type(4))) float v4f;
typedef __attribute__((ext_vector_type(8))) float v8f;

#define B_   32
#define C_   256
#define SQ_  64
#define HW_  4096   // 64*64

// ---------------------------------------------------------------------------
// Kernel 1: per-(b,c) spatial mean. One block per plane (8192 planes),
// 256 threads * 4 float4 = 4096 floats. Default (RT) loads keep x in L2
// for kernel 3 (x = 128 MB < 192 MB L2).
// ---------------------------------------------------------------------------
__global__ __launch_bounds__(256) void se_mean_kernel(
    const float* __restrict__ x, float* __restrict__ m) {
  const int bc = blockIdx.x;
  const v4f* p = (const v4f*)(x + (size_t)bc * HW_);
  const int t = threadIdx.x;

  v4f v0 = p[t];
  v4f v1 = p[t + 256];
  v4f v2 = p[t + 512];
  v4f v3 = p[t + 768];
  float s = (v0.x + v0.y + v0.z + v0.w) + (v1.x + v1.y + v1.z + v1.w) +
            (v2.x + v2.y + v2.z + v2.w) + (v3.x + v3.y + v3.z + v3.w);

  // wave32 reduction
  for (int off = 16; off > 0; off >>= 1) s += __shfl_down(s, off, 32);

  __shared__ float red[8];
  if ((t & 31) == 0) red[t >> 5] = s;
  __syncthreads();
  if (t == 0) {
    float tot = 0.f;
#pragma unroll
    for (int i = 0; i < 8; ++i) tot += red[i];
    m[bc] = tot * (1.0f / (float)HW_);
  }
}

// ---------------------------------------------------------------------------
// Kernel 2: gate = hardsigmoid(relu(m @ W1^T + b1) @ W2^T + b2)
// Single block, 8 waves, fp32 WMMA 16x16x4 tiles (exact fp32 math).
//
// f32 A 16x4 layout (ISA 7.12.2): lane L holds row M=L&15;
//   lanes 0-15: VGPR0=K0, VGPR1=K1; lanes 16-31: VGPR0=K2, VGPR1=K3
//   -> a[v] = A[L&15][kbase + 2*(L>>4) + v]
// B 4x16 is symmetric with N across lanes: b[v] = Bmat[2*(L>>4)+v][L&15].
// Since B tile = W^T, b[v] = W[ntile*16 + (L&15)][kbase + 2*(L>>4) + v],
// i.e. identical row-major gather as A.
// C/D 16x16: VGPR r -> row = r + 8*(L>>4), col = L&15.
// ---------------------------------------------------------------------------
__global__ __launch_bounds__(256) void se_gate_kernel(
    const float* __restrict__ m,    // [32][256]
    const float* __restrict__ W1,   // [64][256]
    const float* __restrict__ b1,   // [64]
    const float* __restrict__ W2,   // [256][64]
    const float* __restrict__ b2,   // [256]
    float* __restrict__ gate) {     // [32][256]
  __shared__ float s_lds[B_][SQ_];  // relu(fc1) result

  const int lane  = threadIdx.x & 31;
  const int wave  = threadIdx.x >> 5;   // 0..7
  const int row16 = lane & 15;
  const int khalf = (lane >> 4) << 1;   // 0 or 2

  // ---- GEMM1: S[32][64] = m @ W1^T ; one 16x16 tile per wave ----
  {
    const int mt = wave >> 2;           // 0..1
    const int nt = wave & 3;            // 0..3
    const float* arow = m  + (size_t)(mt * 16 + row16) * C_;
    const float* brow = W1 + (size_t)(nt * 16 + row16) * C_;
    v8f acc = {};
    for (int k0 = 0; k0 < C_; k0 += 4) {
      v2f a, b;
      a.x = arow[k0 + khalf + 0];
      a.y = arow[k0 + khalf + 1];
      b.x = brow[k0 + khalf + 0];
      b.y = brow[k0 + khalf + 1];
      acc = __builtin_amdgcn_wmma_f32_16x16x4_f32(
          false, a, false, b, (short)0, acc, false, false);
    }
    const int col   = nt * 16 + row16;
    const int rbase = mt * 16 + ((lane >> 4) << 3);
    const float bias = b1[col];
#pragma unroll
    for (int r = 0; r < 8; ++r) {
      float v = acc[r] + bias;
      s_lds[rbase + r][col] = v > 0.f ? v : 0.f;   // relu
    }
  }
  __syncthreads();

  // ---- GEMM2: T[32][256] = S @ W2^T ; 32 tiles, 4 per wave ----
#pragma unroll
  for (int t = 0; t < 4; ++t) {
    const int idx = wave * 4 + t;
    const int mt = idx >> 4;            // 0..1
    const int nt = idx & 15;            // 0..15
    const float* arow = &s_lds[mt * 16 + row16][0];
    const float* brow = W2 + (size_t)(nt * 16 + row16) * SQ_;
    v8f acc = {};
#pragma unroll
    for (int k0 = 0; k0 < SQ_; k0 += 4) {
      v2f a, b;
      a.x = arow[k0 + khalf + 0];
      a.y = arow[k0 + khalf + 1];
      b.x = brow[k0 + khalf + 0];
      b.y = brow[k0 + khalf + 1];
      acc = __builtin_amdgcn_wmma_f32_16x16x4_f32(
          false, a, false, b, (short)0, acc, false, false);
    }
    const int col   = nt * 16 + row16;
    const int rbase = mt * 16 + ((lane >> 4) << 3);
    const float bias = b2[col];
#pragma unroll
    for (int r = 0; r < 8; ++r) {
      float v = acc[r] + bias;
      v = v * (1.0f / 6.0f) + 0.5f;                 // hardsigmoid
      v = fminf(fmaxf(v, 0.0f), 1.0f);
      gate[(rbase + r) * C_ + col] = v;
    }
  }
}

// ---------------------------------------------------------------------------
// Kernel 3: y = x * gate[b,c]. One block per plane; x read is last-use (NT),
// y store is nontemporal so it streams to HBM without thrashing L2.
// ---------------------------------------------------------------------------
__global__ __launch_bounds__(256) void se_scale_kernel(
    const float* __restrict__ x, const float* __restrict__ gate,
    float* __restrict__ y) {
  const int bc = blockIdx.x;
  const float g = gate[bc];             // uniform -> scalar load + broadcast
  const v4f* px = (const v4f*)(x + (size_t)bc * HW_);
  v4f* py = (v4f*)(y + (size_t)bc * HW_);
  const int t = threadIdx.x;
#pragma unroll
  for (int i = 0; i < 4; ++i) {
    v4f v = __builtin_nontemporal_load(&px[t + i * 256]);
    v.x *= g; v.y *= g; v.z *= g; v.w *= g;
    __builtin_nontemporal_store(v, &py[t + i * 256]);
  }
}

// ---------------------------------------------------------------------------
extern "C" void kernel_launch(void* const* d_in, const int* in_sizes, int n_in,
                              void* d_out, int out_size, void* d_ws, size_t ws_size,
                              hipStream_t stream) {
  const float* x  = (const float*)d_in[0];   // [32,256,64,64]
  const float* W1 = (const float*)d_in[1];   // [64,256]
  const float* b1 = (const float*)d_in[2];   // [64]
  const float* W2 = (const float*)d_in[3];   // [256,64]
  const float* b2 = (const float*)d_in[4];   // [256]
  float* y = (float*)d_out;

  float* m    = (float*)d_ws;                // [32*256]
  float* gate = m + (B_ * C_);               // [32*256]

  const int nplanes = B_ * C_;               // 8192
  se_mean_kernel<<<nplanes, 256, 0, stream>>>(x, m);
  se_gate_kernel<<<1, 256, 0, stream>>>(m, W1, b1, W2, b2, gate);
  se_scale_kernel<<<nplanes, 256, 0, stream>>>(x, gate, y);
}